// penalty_focal_arcloss_1211180778084
// MI455X (gfx1250) — compile-verified
//
#include <hip/hip_runtime.h>
#include <math.h>

typedef __attribute__((ext_vector_type(2))) float v2f;
typedef __attribute__((ext_vector_type(8))) float v8f;

#define B_ROWS 131072
#define C_CLS  128
#define E_DIM  128
#define NBLK   512

#define WAVES_PER_WG  4
#define ROWS_PER_WAVE 16
#define ROWS_PER_WG   (WAVES_PER_WG * ROWS_PER_WAVE)
#define LDA 132   // padded LDS row stride (floats) -> conflict-free wmma-A reads

__device__ __constant__ float kS    = 100.0f;
__device__ __constant__ float kCOSM = 0.5403023058681398f;   // cos(1.0)
__device__ __constant__ float kSINM = 0.8414709848078965f;   // sin(1.0)
__device__ __constant__ float kTH   = -0.5403023058681398f;  // cos(pi-1)
__device__ __constant__ float kMM   = 0.8414709848078965f;   // sin(pi-1)*1.0

// ---------------- Kernel A: row-normalize weight [C,E] -> ws ----------------
__global__ __launch_bounds__(32) void norm_weight_kernel(const float* __restrict__ w,
                                                         float* __restrict__ wn) {
  const int row  = blockIdx.x;          // 0..127, one wave per row
  const int lane = threadIdx.x;         // 0..31 (wave32)
  const float4 v = reinterpret_cast<const float4*>(w + (size_t)row * E_DIM)[lane];
  float ss = v.x * v.x + v.y * v.y + v.z * v.z + v.w * v.w;
  for (int m = 1; m <= 16; m <<= 1) ss += __shfl_xor(ss, m, 32);
  const float inv = 1.0f / fmaxf(sqrtf(ss), 1e-12f);
  float4 o;
  o.x = v.x * inv; o.y = v.y * inv; o.z = v.z * inv; o.w = v.w * inv;
  reinterpret_cast<float4*>(wn + (size_t)row * E_DIM)[lane] = o;
}

// ---------------- Kernel B: fused GEMM (WMMA f32) + arc margin + log-softmax ----------------
__global__ __launch_bounds__(WAVES_PER_WG * 32) void arc_main_kernel(
    const float* __restrict__ predict, const float* __restrict__ target,
    const float* __restrict__ wn, float* __restrict__ partial) {
  __shared__ float a_tiles[WAVES_PER_WG * ROWS_PER_WAVE * LDA];
  __shared__ float inv_n[WAVES_PER_WG][ROWS_PER_WAVE];
  __shared__ int   lbl[WAVES_PER_WG][ROWS_PER_WAVE];
  __shared__ float wg_part[WAVES_PER_WG];

  const int lane    = threadIdx.x & 31;
  const int wave    = threadIdx.x >> 5;
  const int l16     = lane & 15;
  const int halfSel = lane >> 4;        // 0: rows j, 1: rows j+8 in C/D layout
  const int koff    = halfSel * 2;      // A/B K sub-offset per ISA 16x4 f32 layout

  float* a_lds = &a_tiles[wave * ROWS_PER_WAVE * LDA];
  float nll_acc = 0.0f;

  const int nRowBlocks = B_ROWS / ROWS_PER_WG;   // 2048
  for (int rb = blockIdx.x; rb < nRowBlocks; rb += gridDim.x) {
    const int rowBase = rb * ROWS_PER_WG + wave * ROWS_PER_WAVE;

    // ---- stage predict tile (coalesced) + row norms + labels ----
    for (int r = 0; r < ROWS_PER_WAVE; ++r) {
      const float4 p = reinterpret_cast<const float4*>(
          predict + (size_t)(rowBase + r) * E_DIM)[lane];
      float ss = p.x * p.x + p.y * p.y + p.z * p.z + p.w * p.w;
      for (int m = 1; m <= 16; m <<= 1) ss += __shfl_xor(ss, m, 32);

      const float4 t = reinterpret_cast<const float4*>(
          target + (size_t)(rowBase + r) * C_CLS)[lane];
      float lf = t.x * (float)(4 * lane)     + t.y * (float)(4 * lane + 1) +
                 t.z * (float)(4 * lane + 2) + t.w * (float)(4 * lane + 3);
      for (int m = 1; m <= 16; m <<= 1) lf += __shfl_xor(lf, m, 32);

      reinterpret_cast<float4*>(a_lds + r * LDA)[lane] = p;
      if (lane == 0) {
        inv_n[wave][r] = 1.0f / fmaxf(sqrtf(ss), 1e-12f);
        lbl[wave][r]   = (int)(lf + 0.5f);
      }
    }

    // ---- GEMM: 16 rows x 128 cols, K=128 via 32 x V_WMMA_F32_16X16X4_F32 ----
    v8f acc[8];
#pragma unroll
    for (int t = 0; t < 8; ++t) acc[t] = (v8f){0, 0, 0, 0, 0, 0, 0, 0};

    const float* aptr = a_lds + l16 * LDA + koff;          // LDS, ds_load_b64
    const float* bptr = wn + (size_t)l16 * E_DIM + koff;   // global, L2/WGP$ resident

#pragma unroll 4
    for (int kk = 0; kk < 32; ++kk) {
      const v2f a = *reinterpret_cast<const v2f*>(aptr + 4 * kk);
#pragma unroll
      for (int t = 0; t < 8; ++t) {
        const v2f b = *reinterpret_cast<const v2f*>(bptr + (size_t)t * 16 * E_DIM + 4 * kk);
        acc[t] = __builtin_amdgcn_wmma_f32_16x16x4_f32(
            false, a, false, b, (short)0, acc[t], false, false);
      }
    }

    // ---- epilogue: scale -> cosine, arc margin at label, log-softmax, NLL ----
#pragma unroll
    for (int j = 0; j < 8; ++j) {
      const int   row  = j + 8 * halfSel;
      const float invn = inv_n[wave][row];
      const int   lab  = lbl[wave][row];
      float lg[8];
      float mx  = -1e30f;
      float sel = 0.0f;
#pragma unroll
      for (int t = 0; t < 8; ++t) {
        const float c   = acc[t][j] * invn;
        const int   col = 16 * t + l16;
        float l;
        if (col == lab) {
          const float sine = sqrtf(fmaxf(1.0f - c * c, 0.0f));
          float phi = c * kCOSM - sine * kSINM;
          phi = (c > kTH) ? phi : (c - kMM);
          l = kS * phi;
          sel += l;
        } else {
          l = kS * c;
        }
        lg[t] = l;
        mx = fmaxf(mx, l);
      }
      // half-wave (16-lane) butterfly reductions: one row per half-wave
      for (int m = 1; m <= 8; m <<= 1) mx = fmaxf(mx, __shfl_xor(mx, m, 32));
      float se = 0.0f;
#pragma unroll
      for (int t = 0; t < 8; ++t) se += expf(lg[t] - mx);
      for (int m = 1; m <= 8; m <<= 1) se  += __shfl_xor(se, m, 32);
      for (int m = 1; m <= 8; m <<= 1) sel += __shfl_xor(sel, m, 32);
      nll_acc += (mx + logf(se)) - sel;   // replicated across the 16 lanes of the half
    }
  }

  // combine the two half-waves (rows 0-7 on lanes<16, rows 8-15 on lanes>=16)
  float tot = nll_acc + __shfl_xor(nll_acc, 16, 32);
  if (lane == 0) wg_part[wave] = tot;
  __syncthreads();
  if (threadIdx.x == 0)
    partial[blockIdx.x] = wg_part[0] + wg_part[1] + wg_part[2] + wg_part[3];
}

// ---------------- Kernel C: deterministic tree reduction -> mean ----------------
__global__ __launch_bounds__(NBLK) void reduce_kernel(const float* __restrict__ partial,
                                                      float* __restrict__ out) {
  __shared__ float s[NBLK];
  const int t = threadIdx.x;
  s[t] = partial[t];
  __syncthreads();
  for (int off = NBLK / 2; off > 0; off >>= 1) {
    if (t < off) s[t] += s[t + off];
    __syncthreads();
  }
  if (t == 0) out[0] = s[0] * (1.0f / (float)B_ROWS);
}

extern "C" void kernel_launch(void* const* d_in, const int* in_sizes, int n_in,
                              void* d_out, int out_size, void* d_ws, size_t ws_size,
                              hipStream_t stream) {
  (void)in_sizes; (void)n_in; (void)out_size; (void)ws_size;
  const float* predict = (const float*)d_in[0];
  const float* target  = (const float*)d_in[1];
  const float* weight  = (const float*)d_in[2];

  float* wn      = (float*)d_ws;                 // 128*128 floats = 64 KB
  float* partial = wn + C_CLS * E_DIM;           // NBLK floats
  float* out     = (float*)d_out;

  norm_weight_kernel<<<C_CLS, 32, 0, stream>>>(weight, wn);
  arc_main_kernel<<<NBLK, WAVES_PER_WG * 32, 0, stream>>>(predict, target, wn, partial);
  reduce_kernel<<<1, NBLK, 0, stream>>>(partial, out);
}